// ModuleRenderScatter_72404558676115
// MI455X (gfx1250) — compile-verified
//
#include <hip/hip_runtime.h>

// Bokeh splat (gather form) for MI455X / gfx1250.
// - 16x16 output tile per 256-thread block (8 wave32s).
// - 34x34 halo staged in LDS: float4 RGBA plane (filled by gfx1250 async
//   global->LDS DMA, ASYNCcnt) + float2 {d, 1/(d^2+0.2)} plane.
// - Per (pixel, offset) pair: 1 ds_load_b128 + 1 ds_load_b64, window test is
//   a single compare (r >= m-1), sigmoid via v_exp_f32 + v_rcp_f32,
//   4 accumulation FMAs, 1 cmp + cndmask/max for the dilation map.

#define HH   1080
#define WW   1920
#define KR   9               // K = RMAX + 1
#define TILE 16
#define SP   (TILE + 2*KR)   // 34: LDS plane stride
#define NHALO (SP * SP)      // 1156
#define CEXP 11.5415603f     // 8 * log2(e)

__global__ __launch_bounds__(256)
void bokeh_splat_kernel(const float* __restrict__ image,     // [3,H,W]
                        const float* __restrict__ defocus,   // [H,W]
                        float* __restrict__ outBokeh,        // [3,H,W]
                        int*   __restrict__ outDil)          // [H,W]
{
    __shared__ float4 s_rgba[NHALO];    // .x=R .y=G .z=B (.w unused)
    __shared__ float2 s_dinv[NHALO];    // {d, 1/(d^2+0.2)}
    __shared__ float2 s_tab[19 * 19];   // {dist, m-1} per offset

    const int tid = threadIdx.x;
    const int tx0 = blockIdx.x * TILE;
    const int ty0 = blockIdx.y * TILE;

    // Per-offset constants (broadcast table).
    if (tid < 361) {
        int dy = tid / 19 - KR;
        int dx = tid % 19 - KR;
        float dist = sqrtf((float)(dy * dy + dx * dx));
        int m = max(abs(dy), abs(dx));
        s_tab[tid] = make_float2(dist, (float)(m - 1));
    }

    const float* imgR = image;
    const float* imgG = image + HH * WW;
    const float* imgB = image + 2 * HH * WW;

    // Halo staging. Image planes: async global->LDS DMA, coords clamped
    // (halo image values get w==0, so clamping is exact). Defocus plane:
    // d=0, inv=0 outside (kills both weight and dilation contributions).
    for (int i = tid; i < NHALO; i += 256) {
        int ly = i / SP;
        int lx = i - ly * SP;
        int gy = ty0 - KR + ly;
        int gx = tx0 - KR + lx;
        bool valid = (gy >= 0) && (gy < HH) && (gx >= 0) && (gx < WW);
        int cgy = min(max(gy, 0), HH - 1);
        int cgx = min(max(gx, 0), WW - 1);
        unsigned gOff = (unsigned)(cgy * WW + cgx) * 4u;   // byte offset

        // Interleave the three planar channels into one float4 LDS slot:
        // per-lane LDS destinations base+0 / base+4 / base+8.
        unsigned lR = (unsigned)(unsigned long long)&s_rgba[i];
        unsigned lG = lR + 4u;
        unsigned lB = lR + 8u;
        asm volatile("global_load_async_to_lds_b32 %0, %1, %2"
                     :: "v"(lR), "v"(gOff), "s"(imgR) : "memory");
        asm volatile("global_load_async_to_lds_b32 %0, %1, %2"
                     :: "v"(lG), "v"(gOff), "s"(imgG) : "memory");
        asm volatile("global_load_async_to_lds_b32 %0, %1, %2"
                     :: "v"(lB), "v"(gOff), "s"(imgB) : "memory");

        float dv = 0.0f, iv = 0.0f;
        if (valid) {
            float dd = defocus[cgy * WW + cgx];
            dv = dd;
            iv = 1.0f / (dd * dd + 0.2f);
        }
        s_dinv[i] = make_float2(dv, iv);    // one ds_store_b64
    }
    asm volatile("s_wait_asynccnt 0x0" ::: "memory");
    __syncthreads();

    const int ty = tid >> 4;
    const int tx = tid & 15;
    const int px = tx0 + tx;
    const int py = ty0 + ty;
    const int center = (ty + KR) * SP + (tx + KR);

    float accR = 0.0f, accG = 0.0f, accB = 0.0f, accW = 0.0f, dmax = 0.0f;

#pragma unroll 1
    for (int dy = -KR; dy <= KR; ++dy) {
        const int rowoff = center + dy * SP;
        const float2* trow = &s_tab[(dy + KR) * 19];
#pragma unroll
        for (int j = 0; j < 19; ++j) {
            float2 t  = trow[j];                 // {dist, m-1} broadcast b64
            int idx   = rowoff + (j - KR);
            float2 di = s_dinv[idx];             // ds_load_b64
            float4 im = s_rgba[idx];             // ds_load_b128
            float r = fabsf(di.x);
            // sigmoid(8*(r-dist)) = 1 / (1 + exp2(CEXP*(dist-r)))
            float e   = __builtin_amdgcn_exp2f((t.x - r) * CEXP);
            float sgm = __builtin_amdgcn_rcpf(1.0f + e);
            float w   = (r >= t.y) ? sgm * di.y : 0.0f;   // window test
            accR = fmaf(w, im.x, accR);
            accG = fmaf(w, im.y, accG);
            accB = fmaf(w, im.z, accB);
            accW += w;
            // dilation: r >= dist implies the window test (dist >= m > m-1)
            dmax = (r >= t.x) ? fmaxf(dmax, truncf(di.x)) : dmax;
        }
    }

    if (px < WW && py < HH) {
        const int o = py * WW + px;
        float invW = 1.0f / accW;     // accW > 0: the delta=0 term is > 0
        outBokeh[o]               = accR * invW;
        outBokeh[HH * WW + o]     = accG * invW;
        outBokeh[2 * HH * WW + o] = accB * invW;
        float dp = s_dinv[center].x;  // defocus at p
        outDil[o] = (int)fmaxf(dp, dmax);   // trunc(max(d, dil_max)), dmax >= 0
    }
}

extern "C" void kernel_launch(void* const* d_in, const int* in_sizes, int n_in,
                              void* d_out, int out_size, void* d_ws, size_t ws_size,
                              hipStream_t stream) {
    (void)in_sizes; (void)n_in; (void)d_ws; (void)ws_size; (void)out_size;
    const float* image   = (const float*)d_in[0];
    const float* defocus = (const float*)d_in[1];
    float* outBokeh = (float*)d_out;
    int*   outDil   = (int*)(outBokeh + 3 * HH * WW);

    dim3 grid((WW + TILE - 1) / TILE, (HH + TILE - 1) / TILE);
    bokeh_splat_kernel<<<grid, 256, 0, stream>>>(image, defocus, outBokeh, outDil);
}